// GATBase_61229053772380
// MI455X (gfx1250) — compile-verified
//
#include <hip/hip_runtime.h>
#include <hip/hip_bf16.h>
#include <math.h>

// Problem constants (match reference)
#define NNODES 50000
#define NEDGES 800000
#define ETOT   (NEDGES + NNODES)   // edges + self loops
#define FIN    256
#define H1     8
#define C1     32
#define D1     256                 // H1*C1
#define OUT2   2
#define SLOPE  0.2f

#define LDSPAD 4                   // pad A rows to 260 floats: conflict-free + 16B aligned
#define ASTRIDE (FIN + LDSPAD)     // 260

typedef float v2f __attribute__((ext_vector_type(2)));
typedef float v8f __attribute__((ext_vector_type(8)));

// ---------------------------------------------------------------------------
// helpers
// ---------------------------------------------------------------------------
__device__ __forceinline__ void edge_sd(const int* __restrict__ ei, int e, int& s, int& d) {
    if (e < NEDGES) { s = ei[e]; d = ei[NEDGES + e]; }
    else            { s = e - NEDGES; d = s; }       // self loop
}

__device__ __forceinline__ float lrelu(float v) {
    return v > 0.0f ? v : SLOPE * v;
}

// float atomic-max via one hardware integer atomic (no CAS loop).
// Works for any mix of signs given the slot is initialized to -inf.
__device__ __forceinline__ void atomicMaxF(float* addr, float val) {
    if (val >= 0.0f) atomicMax((int*)addr, __float_as_int(val));
    else             atomicMin((unsigned int*)addr, __float_as_uint(val));
}

// ---------------------------------------------------------------------------
// Layer-1 GEMM: xp = x @ W1   (50000x256 @ 256x256), fp32 WMMA 16x16x4.
// One 512-thread block (16 waves) owns a 16x256 output row-block:
//   - cooperatively stage the 16x256 A-panel in LDS once (kills the 16x
//     re-read of x that a wave-per-tile scheme would incur),
//   - wave w computes output tileN = w, reading A from LDS (ds_load_b64,
//     bank-conflict-free via 260-float row stride) and B from L2-resident W1.
// 64 WMMAs per wave over K=256.
// ---------------------------------------------------------------------------
__global__ __launch_bounds__(512) void gemm1_wmma(const float* __restrict__ x,
                                                  const float* __restrict__ W1,
                                                  float* __restrict__ xp) {
    __shared__ float aTile[16 * ASTRIDE];

    const int t     = threadIdx.x;
    const int lane  = t & 31;
    const int wave  = t >> 5;            // 0..15 == tileN
    const int tileM = blockIdx.x;        // 0..3124  (3125*16 == 50000)
    const int half  = lane >> 4;         // 0 | 1
    const int l16   = lane & 15;

    // ---- cooperative A-panel load: 16 rows x 256 cols, 8 floats / thread ----
    {
        const int row = t >> 5;              // t/32: 16 rows, one per wave
        const int col = (t & 31) * 8;        // 8 consecutive floats
        const float4* src = (const float4*)(x + (size_t)(tileM * 16 + row) * FIN + col);
        float4* dst = (float4*)(aTile + row * ASTRIDE + col);
        dst[0] = src[0];                     // ds_store_b128
        dst[1] = src[1];
    }
    __syncthreads();

    const int bcol = wave * 16 + l16;        // B-matrix col for this lane

    v8f acc = {};
#pragma unroll
    for (int k = 0; k < FIN; k += 4) {
        const int kk = k + half * 2;         // K pair this half-wave owns
        // A 16x4 fp32 tile from LDS: lane<16 -> K0,K1 ; lane>=16 -> K2,K3
        v2f a = *(const v2f*)(aTile + l16 * ASTRIDE + kk);   // ds_load_b64
        // B 4x16 fp32 tile (mirrored layout across lane halves), L2-resident
        v2f b;
        b.x = W1[kk * D1 + bcol];
        b.y = W1[(kk + 1) * D1 + bcol];
        acc = __builtin_amdgcn_wmma_f32_16x16x4_f32(
            /*neg_a=*/false, a, /*neg_b=*/false, b,
            /*c_mod=*/(short)0, acc, /*reuse_a=*/false, /*reuse_b=*/false);
    }

    // C/D layout: VGPR r = row r (lanes 0-15) / row 8+r (lanes 16-31)
    const int mrow = tileM * 16 + half * 8;
#pragma unroll
    for (int r = 0; r < 8; ++r)
        xp[(size_t)(mrow + r) * D1 + bcol] = acc[r];
}

// ---------------------------------------------------------------------------
// Per-node attention logits for layer 1: as/ad[n,h] = <xp[n,h,:], att_{src,dst}[h,:]>
// ---------------------------------------------------------------------------
__global__ void alpha1_kernel(const float* __restrict__ xp,
                              const float* __restrict__ a_src,
                              const float* __restrict__ a_dst,
                              float* __restrict__ as, float* __restrict__ ad) {
    int i = blockIdx.x * blockDim.x + threadIdx.x;   // n*H1 + h
    if (i >= NNODES * H1) return;
    int h = i & (H1 - 1);
    int n = i >> 3;
    const float* row = xp + (size_t)n * D1 + h * C1;
    const float* ws  = a_src + h * C1;
    const float* wd  = a_dst + h * C1;
    float s = 0.0f, d = 0.0f;
#pragma unroll
    for (int c = 0; c < C1; ++c) {
        float v = row[c];
        s = fmaf(v, ws[c], s);
        d = fmaf(v, wd[c], d);
    }
    as[i] = s;
    ad[i] = d;
}

// ---------------------------------------------------------------------------
// fill buffer with constant (used for -inf segment-max init)
// ---------------------------------------------------------------------------
__global__ void fill_f32(float* __restrict__ p, float v, int n) {
    int i = blockIdx.x * blockDim.x + threadIdx.x;
    if (i < n) p[i] = v;
}

// ---------------------------------------------------------------------------
// Edge pass 1: segment max of leaky-relu logits, grouped by dst. Thread/edge,
// inner loop over heads.
// ---------------------------------------------------------------------------
template <int NH>
__global__ void edge_max(const int* __restrict__ ei,
                         const float* __restrict__ as,
                         const float* __restrict__ ad,
                         float* __restrict__ m) {
    int e = blockIdx.x * blockDim.x + threadIdx.x;
    if (e >= ETOT) return;
    int s, d;
    edge_sd(ei, e, s, d);
#pragma unroll
    for (int h = 0; h < NH; ++h) {
        float v = lrelu(as[s * NH + h] + ad[d * NH + h]);
        atomicMaxF(&m[d * NH + h], v);
    }
}

// ---------------------------------------------------------------------------
// Edge pass 2: segment sum of exp(e - max), grouped by dst.
// ---------------------------------------------------------------------------
template <int NH>
__global__ void edge_denom(const int* __restrict__ ei,
                           const float* __restrict__ as,
                           const float* __restrict__ ad,
                           const float* __restrict__ m,
                           float* __restrict__ den) {
    int e = blockIdx.x * blockDim.x + threadIdx.x;
    if (e >= ETOT) return;
    int s, d;
    edge_sd(ei, e, s, d);
#pragma unroll
    for (int h = 0; h < NH; ++h) {
        float v = lrelu(as[s * NH + h] + ad[d * NH + h]);
        atomicAdd(&den[d * NH + h], __expf(v - m[d * NH + h]));
    }
}

// ---------------------------------------------------------------------------
// Layer-1 aggregation: out[dst,h,c] += alpha(e,h) * xp[src,h,c].
// One 256-thread block per edge; thread = channel, wave = head (wave32!).
// alpha is wave-uniform -> scalar broadcast loads.
// ---------------------------------------------------------------------------
__global__ __launch_bounds__(256) void agg1(const int* __restrict__ ei,
                                            const float* __restrict__ xp,
                                            const float* __restrict__ as,
                                            const float* __restrict__ ad,
                                            const float* __restrict__ m,
                                            const float* __restrict__ den,
                                            float* __restrict__ out) {
    int e = blockIdx.x;
    int t = threadIdx.x;       // channel 0..255
    int h = t >> 5;            // head == wave id
    int s, d;
    edge_sd(ei, e, s, d);
    float v     = lrelu(as[s * H1 + h] + ad[d * H1 + h]);
    float alpha = __expf(v - m[d * H1 + h]) / den[d * H1 + h];
    atomicAdd(&out[(size_t)d * D1 + t], alpha * xp[(size_t)s * D1 + t]);
}

// ---------------------------------------------------------------------------
// h = relu(agg + b1)
// ---------------------------------------------------------------------------
__global__ void bias_relu(float* __restrict__ hbuf, const float* __restrict__ b) {
    int i = blockIdx.x * blockDim.x + threadIdx.x;
    if (i >= NNODES * D1) return;
    float v = hbuf[i] + b[i & (D1 - 1)];
    hbuf[i] = v > 0.0f ? v : 0.0f;
}

// ---------------------------------------------------------------------------
// Layer-2 projection (256 -> 2) + attention logits, wave per node.
// Only 2 output columns: WMMA would waste 14/16 of the tile, so shuffle-reduce.
// ---------------------------------------------------------------------------
__global__ __launch_bounds__(256) void gemm2_alpha2(const float* __restrict__ h,
                                                    const float* __restrict__ W2,
                                                    const float* __restrict__ aw_src,
                                                    const float* __restrict__ aw_dst,
                                                    float* __restrict__ xp2,
                                                    float* __restrict__ as2,
                                                    float* __restrict__ ad2) {
    int lane = threadIdx.x & 31;
    int n    = blockIdx.x * (blockDim.x >> 5) + (threadIdx.x >> 5);
    if (n >= NNODES) return;
    float s0 = 0.0f, s1 = 0.0f;
    for (int k = lane; k < D1; k += 32) {
        float v = h[(size_t)n * D1 + k];
        s0 = fmaf(v, W2[k * OUT2 + 0], s0);
        s1 = fmaf(v, W2[k * OUT2 + 1], s1);
    }
#pragma unroll
    for (int off = 16; off > 0; off >>= 1) {
        s0 += __shfl_down(s0, off);
        s1 += __shfl_down(s1, off);
    }
    if (lane == 0) {
        xp2[n * OUT2 + 0] = s0;
        xp2[n * OUT2 + 1] = s1;
        as2[n] = fmaf(s0, aw_src[0], s1 * aw_src[1]);
        ad2[n] = fmaf(s0, aw_dst[0], s1 * aw_dst[1]);
    }
}

// ---------------------------------------------------------------------------
// Layer-2 aggregation: thread per edge, 2 channels.
// ---------------------------------------------------------------------------
__global__ void agg2(const int* __restrict__ ei,
                     const float* __restrict__ xp2,
                     const float* __restrict__ as2,
                     const float* __restrict__ ad2,
                     const float* __restrict__ m2,
                     const float* __restrict__ den2,
                     float* __restrict__ out) {
    int e = blockIdx.x * blockDim.x + threadIdx.x;
    if (e >= ETOT) return;
    int s, d;
    edge_sd(ei, e, s, d);
    float v     = lrelu(as2[s] + ad2[d]);
    float alpha = __expf(v - m2[d]) / den2[d];
    atomicAdd(&out[d * OUT2 + 0], alpha * xp2[s * OUT2 + 0]);
    atomicAdd(&out[d * OUT2 + 1], alpha * xp2[s * OUT2 + 1]);
}

__global__ void bias2_add(float* __restrict__ out, const float* __restrict__ b) {
    int i = blockIdx.x * blockDim.x + threadIdx.x;
    if (i < NNODES * OUT2) out[i] += b[i & 1];
}

// ---------------------------------------------------------------------------
// launch
// ---------------------------------------------------------------------------
extern "C" void kernel_launch(void* const* d_in, const int* in_sizes, int n_in,
                              void* d_out, int out_size, void* d_ws, size_t ws_size,
                              hipStream_t stream) {
    const float* x     = (const float*)d_in[0];
    const int*   ei    = (const int*)d_in[1];    // edge_index [2,E]
    const float* W1    = (const float*)d_in[2];
    const float* asw1  = (const float*)d_in[3];
    const float* adw1  = (const float*)d_in[4];
    const float* b1    = (const float*)d_in[5];
    const float* W2    = (const float*)d_in[6];
    const float* asw2  = (const float*)d_in[7];
    const float* adw2  = (const float*)d_in[8];
    const float* b2    = (const float*)d_in[9];
    float*       out   = (float*)d_out;

    // workspace carve-up (floats)
    float* w    = (float*)d_ws;
    float* xp1  = w;                            // N*256
    float* hbuf = xp1  + (size_t)NNODES * D1;   // N*256
    float* as1  = hbuf + (size_t)NNODES * D1;   // N*8
    float* ad1  = as1  + (size_t)NNODES * H1;
    float* m1   = ad1  + (size_t)NNODES * H1;
    float* dn1  = m1   + (size_t)NNODES * H1;
    float* xp2  = dn1  + (size_t)NNODES * H1;   // N*2
    float* as2  = xp2  + (size_t)NNODES * OUT2; // N
    float* ad2  = as2  + (size_t)NNODES;
    float* m2   = ad2  + (size_t)NNODES;
    float* dn2  = m2   + (size_t)NNODES;

    // zero accumulators (graph-capturable stream memsets)
    hipMemsetAsync(hbuf, 0, (size_t)NNODES * D1 * sizeof(float), stream);
    hipMemsetAsync(dn1,  0, (size_t)NNODES * H1 * sizeof(float), stream);
    hipMemsetAsync(dn2,  0, (size_t)NNODES * sizeof(float), stream);
    hipMemsetAsync(out,  0, (size_t)NNODES * OUT2 * sizeof(float), stream);
    // -inf init for segment max
    fill_f32<<<(NNODES * H1 + 255) / 256, 256, 0, stream>>>(m1, -INFINITY, NNODES * H1);
    fill_f32<<<(NNODES + 255) / 256, 256, 0, stream>>>(m2, -INFINITY, NNODES);

    // ---- layer 1 ----
    gemm1_wmma<<<NNODES / 16, 512, 0, stream>>>(x, W1, xp1);  // 3125 row-blocks
    alpha1_kernel<<<(NNODES * H1 + 255) / 256, 256, 0, stream>>>(xp1, asw1, adw1, as1, ad1);
    edge_max<H1><<<(ETOT + 255) / 256, 256, 0, stream>>>(ei, as1, ad1, m1);
    edge_denom<H1><<<(ETOT + 255) / 256, 256, 0, stream>>>(ei, as1, ad1, m1, dn1);
    agg1<<<ETOT, 256, 0, stream>>>(ei, xp1, as1, ad1, m1, dn1, hbuf);
    bias_relu<<<(NNODES * D1 + 255) / 256, 256, 0, stream>>>(hbuf, b1);

    // ---- layer 2 ----
    gemm2_alpha2<<<(NNODES + 7) / 8, 256, 0, stream>>>(hbuf, W2, asw2, adw2, xp2, as2, ad2);
    edge_max<1><<<(ETOT + 255) / 256, 256, 0, stream>>>(ei, as2, ad2, m2);
    edge_denom<1><<<(ETOT + 255) / 256, 256, 0, stream>>>(ei, as2, ad2, m2, dn2);
    agg2<<<(ETOT + 255) / 256, 256, 0, stream>>>(ei, xp2, as2, ad2, m2, dn2, out);
    bias2_add<<<(NNODES * OUT2 + 255) / 256, 256, 0, stream>>>(out, b2);
}